// SAE_Model_56461640073717
// MI455X (gfx1250) — compile-verified
//
#include <hip/hip_runtime.h>
#include <math.h>

typedef __attribute__((ext_vector_type(2))) float v2f;
typedef __attribute__((ext_vector_type(8))) float v8f;

#define N_TOK 512
#define D_IN  4096
#define A_DIM 24576
#define TOPK  32

// ---- CDNA5 async global->LDS path (ASYNCcnt), with safe fallback ----------
#if __has_builtin(__builtin_amdgcn_global_load_async_to_lds_b128)
#define HAS_ASYNC_LDS 1
#else
#define HAS_ASYNC_LDS 0
#endif

// Parameter types per hipcc diagnostic: (int4 __device__*, int4 __shared__*, imm, imm)
typedef int vi4 __attribute__((vector_size(16)));
typedef __attribute__((address_space(1))) vi4 GVi4;  // global int4
typedef __attribute__((address_space(3))) vi4 LVi4;  // LDS int4

// Copy 16 bytes global -> LDS. Async variant bypasses VGPR staging and is
// tracked with ASYNCcnt.
__device__ __forceinline__ void copy16_g2l(const float* __restrict__ g,
                                           float* l) {
#if HAS_ASYNC_LDS
    __builtin_amdgcn_global_load_async_to_lds_b128((GVi4*)g, (LVi4*)l, 0, 0);
#else
    const float4 v = *reinterpret_cast<const float4*>(g);
    l[0] = v.x; l[1] = v.y; l[2] = v.z; l[3] = v.w;
#endif
}

template <int N>
__device__ __forceinline__ void wait_async() {
#if HAS_ASYNC_LDS
#if __has_builtin(__builtin_amdgcn_s_wait_asynccnt)
    __builtin_amdgcn_s_wait_asynccnt(N);
#else
    asm volatile("s_wait_asynccnt %0" ::"n"(N));
#endif
#endif
}

// ---------------------------------------------------------------------------
// Kernel 1: z = relu(x @ W_enc + b_enc)      [512 x 24576]
// grid (4,192): blockIdx.x = row-block (128 rows), blockIdx.y = col-block
// (128 cols). 8 waves: 2 along M x 4 along N; each wave = 64x32 = 4x2 WMMA
// tiles of V_WMMA_F32_16X16X4_F32. K staged through double-buffered LDS in
// chunks of 16 using async global->LDS loads (software pipeline).
// ---------------------------------------------------------------------------
__global__ __launch_bounds__(256)
void sae_encode(const float* __restrict__ x, const float* __restrict__ We,
                const float* __restrict__ be, float* __restrict__ z)
{
    __shared__ float Xs[2][128 * 20];   // 128 rows x 16 K, padded stride 20
    __shared__ float Ws[2][16 * 136];   // 16 K x 128 cols, padded stride 136

    const int tid   = threadIdx.x;
    const int lane  = tid & 31;
    const int wid   = tid >> 5;
    const int hf    = (lane >> 4) & 1;  // 0 = lanes 0-15, 1 = lanes 16-31
    const int l16   = lane & 15;
    const int waveM = wid >> 2;         // 0..1
    const int waveN = wid & 3;          // 0..3
    const int row0  = blockIdx.x * 128;
    const int col0  = blockIdx.y * 128;

    // Per-thread staging assignments (2 float4 for X, 2 float4 for W per chunk)
    const int xr0 = (tid) >> 2,        xc0 = (tid) & 3;
    const int xr1 = (tid + 256) >> 2,  xc1 = (tid + 256) & 3;
    const int wr0 = (tid) >> 5,        wc0 = (tid) & 31;
    const int wr1 = (tid + 256) >> 5,  wc1 = (tid + 256) & 31;

    auto stage_chunk = [&](int k0, int buf) {
        copy16_g2l(x + (size_t)(row0 + xr0) * D_IN + k0 + xc0 * 4,
                   &Xs[buf][xr0 * 20 + xc0 * 4]);
        copy16_g2l(x + (size_t)(row0 + xr1) * D_IN + k0 + xc1 * 4,
                   &Xs[buf][xr1 * 20 + xc1 * 4]);
        copy16_g2l(We + (size_t)(k0 + wr0) * A_DIM + col0 + wc0 * 4,
                   &Ws[buf][wr0 * 136 + wc0 * 4]);
        copy16_g2l(We + (size_t)(k0 + wr1) * A_DIM + col0 + wc1 * 4,
                   &Ws[buf][wr1 * 136 + wc1 * 4]);
    };

    const v8f vzero = {0.f, 0.f, 0.f, 0.f, 0.f, 0.f, 0.f, 0.f};
    v8f acc[4][2];
#pragma unroll
    for (int mt = 0; mt < 4; ++mt)
#pragma unroll
        for (int nt = 0; nt < 2; ++nt) acc[mt][nt] = vzero;

    const int NCH = D_IN / 16;  // 256 chunks
    stage_chunk(0, 0);

    for (int ch = 0; ch < NCH; ++ch) {
        const int buf = ch & 1;
        if (ch + 1 < NCH) {
            stage_chunk((ch + 1) * 16, buf ^ 1);
            wait_async<4>();   // my 4 loads for chunk `ch` have landed in LDS
        } else {
            wait_async<0>();
        }
        __syncthreads();       // everyone's chunk-`ch` data visible

#pragma unroll
        for (int kk = 0; kk < 16; kk += 4) {
            const int ko = kk + 2 * hf;  // lanes 0-15 -> K0,K1 ; 16-31 -> K2,K3
            v2f af[4], bf[2];
#pragma unroll
            for (int mt = 0; mt < 4; ++mt) {
                int r = waveM * 64 + mt * 16 + l16;
                af[mt].x = Xs[buf][r * 20 + ko];
                af[mt].y = Xs[buf][r * 20 + ko + 1];
            }
#pragma unroll
            for (int nt = 0; nt < 2; ++nt) {
                int c = waveN * 32 + nt * 16 + l16;
                bf[nt].x = Ws[buf][ko * 136 + c];
                bf[nt].y = Ws[buf][(ko + 1) * 136 + c];
            }
#pragma unroll
            for (int mt = 0; mt < 4; ++mt)
#pragma unroll
                for (int nt = 0; nt < 2; ++nt)
                    acc[mt][nt] = __builtin_amdgcn_wmma_f32_16x16x4_f32(
                        false, af[mt], false, bf[nt], (short)0, acc[mt][nt],
                        false, false);
        }
        __syncthreads();       // done reading `buf` before it is re-staged
    }

    // epilogue: + b_enc, relu, store z
#pragma unroll
    for (int mt = 0; mt < 4; ++mt) {
#pragma unroll
        for (int nt = 0; nt < 2; ++nt) {
            int gcol = col0 + waveN * 32 + nt * 16 + l16;
            float bias = be[gcol];
#pragma unroll
            for (int r = 0; r < 8; ++r) {
                int grow = row0 + waveM * 64 + mt * 16 + r + 8 * hf;
                float v = acc[mt][nt][r] + bias;
                z[(size_t)grow * A_DIM + gcol] = fmaxf(v, 0.0f);
            }
        }
    }
}

// ---------------------------------------------------------------------------
// Kernel 2: per-row top-32 of z (val desc, idx asc tie-break, matching
// jax.lax.top_k). One block (256 threads) per row; iterative argmax.
// ---------------------------------------------------------------------------
__global__ __launch_bounds__(256)
void sae_topk(const float* __restrict__ z, float* __restrict__ tv,
              int* __restrict__ ti)
{
    const int row = blockIdx.x;
    const int tid = threadIdx.x;
    const float* zr = z + (size_t)row * A_DIM;

    __shared__ float sv[256];
    __shared__ int   si[256];
    __shared__ float bestV;
    __shared__ int   bestI;

    float prevV = __builtin_inff();
    int   prevI = -1;

    for (int it = 0; it < TOPK; ++it) {
        float mv = -__builtin_inff();
        int   mi = 0x7fffffff;
        for (int j = tid; j < A_DIM; j += 256) {
            float v = zr[j];
            bool elig = (v < prevV) || (v == prevV && j > prevI);
            if (elig && (v > mv || (v == mv && j < mi))) { mv = v; mi = j; }
        }
        sv[tid] = mv; si[tid] = mi;
        __syncthreads();
        for (int s = 128; s > 0; s >>= 1) {
            if (tid < s) {
                float ov = sv[tid + s]; int oi = si[tid + s];
                if (ov > sv[tid] || (ov == sv[tid] && oi < si[tid])) {
                    sv[tid] = ov; si[tid] = oi;
                }
            }
            __syncthreads();
        }
        if (tid == 0) {
            bestV = sv[0]; bestI = si[0];
            tv[row * TOPK + it] = sv[0];
            ti[row * TOPK + it] = si[0];
        }
        __syncthreads();
        prevV = bestV; prevI = bestI;
        __syncthreads();
    }
}

// ---------------------------------------------------------------------------
// Kernel 3: sparse decode (32 W_dec rows per token) + cosine/sigmoid scale +
// residual. One block per row; thread t owns D elements t, t+256, ...
// ---------------------------------------------------------------------------
__global__ __launch_bounds__(256)
void sae_decode(const float* __restrict__ x, const float* __restrict__ Wd,
                const float* __restrict__ bd, const float* __restrict__ tv,
                const int* __restrict__ ti, float* __restrict__ out)
{
    const int row = blockIdx.x;
    const int tid = threadIdx.x;
    const float* xr = x + (size_t)row * D_IN;

    float acc[16];
#pragma unroll
    for (int e = 0; e < 16; ++e) acc[e] = bd[tid + e * 256];

    for (int j = 0; j < TOPK; ++j) {
        float v = tv[row * TOPK + j];
        int   id = ti[row * TOPK + j];
        const float* w = Wd + (size_t)id * D_IN;
#pragma unroll
        for (int e = 0; e < 16; ++e) acc[e] += v * w[tid + e * 256];
    }

    float xv[16];
    float dot = 0.f, nx2 = 0.f, nd2 = 0.f;
#pragma unroll
    for (int e = 0; e < 16; ++e) {
        xv[e] = xr[tid + e * 256];
        dot += xv[e] * acc[e];
        nx2 += xv[e] * xv[e];
        nd2 += acc[e] * acc[e];
    }

    __shared__ float rd[256], rx[256], rn[256];
    __shared__ float sScale;
    rd[tid] = dot; rx[tid] = nx2; rn[tid] = nd2;
    __syncthreads();
    for (int s = 128; s > 0; s >>= 1) {
        if (tid < s) {
            rd[tid] += rd[tid + s];
            rx[tid] += rx[tid + s];
            rn[tid] += rn[tid + s];
        }
        __syncthreads();
    }
    if (tid == 0) {
        float nx = sqrtf(rx[0]); if (nx < 1e-8f) nx = 1e-8f;
        float nd = sqrtf(rn[0]); if (nd < 1e-8f) nd = 1e-8f;
        float sim = rd[0] / (nx * nd);
        sScale = 160.0f / (1.0f + expf(-sim));
    }
    __syncthreads();
    const float sc = sScale;
#pragma unroll
    for (int e = 0; e < 16; ++e)
        out[(size_t)row * D_IN + tid + e * 256] = xv[e] + sc * acc[e];
}

// ---------------------------------------------------------------------------
extern "C" void kernel_launch(void* const* d_in, const int* in_sizes, int n_in,
                              void* d_out, int out_size, void* d_ws, size_t ws_size,
                              hipStream_t stream)
{
    const float* acts  = (const float*)d_in[0];  // [64,8,4096]
    const float* W_enc = (const float*)d_in[1];  // [4096,24576]
    const float* b_enc = (const float*)d_in[2];  // [24576]
    const float* W_dec = (const float*)d_in[3];  // [24576,4096]
    const float* b_dec = (const float*)d_in[4];  // [4096]
    // d_in[5] = top_k scalar (32), compile-time constant here

    float* z  = (float*)d_ws;                       // 512*24576 floats (48 MB)
    float* tv = z + (size_t)N_TOK * A_DIM;          // 512*32 floats
    int*   ti = (int*)(tv + (size_t)N_TOK * TOPK);  // 512*32 ints
    float* out = (float*)d_out;

    // Encode: grid.x = row-blocks (4) fastest so blocks sharing a W_enc
    // column panel are launched adjacently -> L2 reuse of the 402 MB matrix.
    sae_encode<<<dim3(4, 192), 256, 0, stream>>>(acts, W_enc, b_enc, z);
    sae_topk<<<N_TOK, 256, 0, stream>>>(z, tv, ti);
    sae_decode<<<N_TOK, 256, 0, stream>>>(acts, W_dec, b_dec, tv, ti, out);
}